// MyLoss_17463337025647
// MI455X (gfx1250) — compile-verified
//
#include <hip/hip_runtime.h>
#include <cstdint>

#define EPSF 1e-6f
#define NP 20
#define NL 12
#define LN2F 0.69314718055994530942f

constexpr int BT = 64;        // batches per block (2 waves of 32)
constexpr int PSTRIDE = 84;   // dwords per batch for preds in LDS (20*4 padded; 16B aligned)
constexpr int LSTRIDE = 52;   // dwords per batch for labels in LDS (12*4 padded; 16B aligned)

__global__ __launch_bounds__(BT)
void pairloss_kernel(const float* __restrict__ pred,
                     const float* __restrict__ label,
                     float* __restrict__ out, int B)
{
    __shared__ __align__(16) float sPred[BT * PSTRIDE]; // ~21.5 KB
    __shared__ __align__(16) float sLab [BT * LSTRIDE]; // ~13.3 KB

    const int tid = threadIdx.x;
    const long long b0 = (long long)blockIdx.x * BT;
    long long rem = (long long)B - b0;
    const int bt = rem < (long long)BT ? (int)rem : BT;

    const uint32_t predLds = (uint32_t)(uintptr_t)&sPred[0];
    const uint32_t labLds  = (uint32_t)(uintptr_t)&sLab[0];

    // ---- CDNA5 async global -> LDS staging (coalesced b128, ASYNCcnt) ----
    {
        const uint64_t gp = (uint64_t)(uintptr_t)(pred + (size_t)b0 * (NP * 4));
        for (int c = tid; c < bt * NP; c += BT) {
            int t = c / NP, j = c - t * NP;            // batch-in-block, pred index
            uint32_t lds = predLds + (uint32_t)(t * PSTRIDE + j * 4) * 4u;
            uint64_t ga  = gp + (uint64_t)c * 16u;
            asm volatile("global_load_async_to_lds_b128 %0, %1, off"
                         :: "v"(lds), "v"(ga) : "memory");
        }
        const uint64_t gl = (uint64_t)(uintptr_t)(label + (size_t)b0 * (NL * 4));
        for (int c = tid; c < bt * NL; c += BT) {
            int t = c / NL, j = c - t * NL;            // batch-in-block, label index
            uint32_t lds = labLds + (uint32_t)(t * LSTRIDE + j * 4) * 4u;
            uint64_t ga  = gl + (uint64_t)c * 16u;
            asm volatile("global_load_async_to_lds_b128 %0, %1, off"
                         :: "v"(lds), "v"(ga) : "memory");
        }
    }
    asm volatile("s_wait_asynccnt 0" ::: "memory"); // this wave's async copies done
    __syncthreads();                                // all waves' LDS writes visible

    if (tid >= bt) return;

    // ---- labels into registers ----
    const float* lb = &sLab[tid * LSTRIDE];
    float lx[NL], ly[NL], lr[NL];
    #pragma unroll
    for (int l = 0; l < NL; ++l) {
        lx[l] = lb[l * 4 + 0];
        ly[l] = lb[l * 4 + 1];
        lr[l] = lb[l * 4 + 2];
    }

    // best2[l] tracks 2x the reference cost: sqrt(d2) + |dr|.
    // x0.5 is an exact power-of-2 scale, so ordering/ties/values match the
    // reference 0.5*dist + 0.5*rdiff exactly after the final 0.5 factor.
    float best2[NL]; int bidx[NL];
    #pragma unroll
    for (int l = 0; l < NL; ++l) { best2[l] = 3.0e38f; bidx[l] = 0; }

    const float* pp = &sPred[tid * PSTRIDE];
    #pragma unroll 4
    for (int p = 0; p < NP; ++p) {
        const float4 pv = *(const float4*)(pp + p * 4);
        #pragma unroll
        for (int l = 0; l < NL; ++l) {
            float dx = lx[l] - pv.x;
            float dy = ly[l] - pv.y;
            float d  = __builtin_amdgcn_sqrtf(__builtin_fmaf(dy, dy, dx * dx)); // raw v_sqrt_f32
            float rd = lr[l] - pv.z;
            float c  = d + __builtin_fabsf(rd);   // |.| is a free src modifier
            if (c < best2[l]) { best2[l] = c; bidx[l] = p; }
        }
    }

    // ---- pair loss: 0.5*sum(best2) - ln2 * sum(log2(w+eps)) ----
    unsigned mask = 0u;
    float bsum = 0.0f, plog2 = 0.0f;
    #pragma unroll
    for (int l = 0; l < NL; ++l) {
        int i = bidx[l];
        mask |= (1u << i);
        float w = pp[i * 4 + 3];
        bsum  += best2[l];
        plog2 += __builtin_amdgcn_logf(w + EPSF);   // raw v_log_f32 (log2)
    }
    float lp = 0.5f * bsum - LN2F * plog2;

    // ---- unpaired penalty: sum_{unmatched} (-ln(1-w+eps) + 0.5*r) * 0.5 ----
    float luA = 0.0f, luB = 0.0f;   // sum log2(1-w+eps), sum r over unmatched
    #pragma unroll 4
    for (int p = 0; p < NP; ++p) {
        const float2 rw = *(const float2*)(pp + p * 4 + 2);   // (r, w), 8B aligned
        float t = __builtin_amdgcn_logf(1.0f - rw.y + EPSF);
        bool matched = (mask >> p) & 1u;
        luA += matched ? 0.0f : t;
        luB += matched ? 0.0f : rw.x;
    }
    float lu = (-0.5f * LN2F) * luA + 0.25f * luB;

    out[b0 + tid] = lp * (1.0f / (float)NL) + lu * (1.0f / (float)(NP - NL));
}

extern "C" void kernel_launch(void* const* d_in, const int* in_sizes, int n_in,
                              void* d_out, int out_size, void* d_ws, size_t ws_size,
                              hipStream_t stream) {
    const float* pred  = (const float*)d_in[0];   // (B, 20, 4) f32
    const float* label = (const float*)d_in[1];   // (B, 12, 4) f32
    float* out = (float*)d_out;                   // (B,) f32
    int B = in_sizes[0] / (NP * 4);
    int grid = (B + BT - 1) / BT;
    pairloss_kernel<<<grid, BT, 0, stream>>>(pred, label, out, B);
}